// ExtractModel_69157563400995
// MI455X (gfx1250) — compile-verified
//
#include <hip/hip_runtime.h>
#include <hip/hip_bf16.h>
#include <float.h>
#include <math.h>

// Problem constants (from reference)
#define B_      8
#define L_      64
#define NT_     8000
#define U_      64
#define G_      6
#define NF_     512
#define D_      256
#define MSL_    10
#define MTL_    10
#define MIN_WL_ 4
#define LEN_E_  7
#define BIG_    99.9f
#define THRESH_ 0.05f
#define NS_     (B_ * L_)

typedef __attribute__((ext_vector_type(2))) float v2f;
typedef __attribute__((ext_vector_type(8))) float v8f;

// ---------------------------------------------------------------------------
// Kernel 1: word_repr / unit_repr gather-sum + reciprocal row norms.
// One wave32 per row, each lane owns 8 contiguous floats of D=256.
// Stores rnx = 1/(||row||+1e-8) so the GEMM epilogue is multiply-only.
// ---------------------------------------------------------------------------
__global__ void repr_kernel(const float* __restrict__ emb,
                            const int* __restrict__ feat,      // B*L*G
                            const int* __restrict__ lengths,   // B
                            const int* __restrict__ ufeat,     // U*G
                            float* __restrict__ word_repr,     // NS*D
                            float* __restrict__ unit_repr,     // U*D
                            float* __restrict__ rnx,           // NS (reciprocal)
                            float* __restrict__ rny)           // U  (reciprocal)
{
    const int wave = threadIdx.x >> 5;
    const int lane = threadIdx.x & 31;
    const int row  = blockIdx.x * 8 + wave;
    if (row >= NS_ + U_) return;

    const int d0 = lane * 8;
    float acc[8];
#pragma unroll
    for (int i = 0; i < 8; ++i) acc[i] = 0.0f;

    const int* idxp;
    float* outp;
    bool mask = false;
    if (row < NS_) {
        const int b = row >> 6, l = row & 63;
        mask = (l >= lengths[b]);
        idxp = feat + row * G_;
        outp = word_repr + row * D_;
    } else {
        idxp = ufeat + (row - NS_) * G_;
        outp = unit_repr + (row - NS_) * D_;
    }

#pragma unroll
    for (int g = 0; g < G_; ++g) {
        const float* e = emb + idxp[g] * D_ + d0;
#pragma unroll
        for (int i = 0; i < 8; ++i) acc[i] += e[i];
    }

    float ss = 0.0f;
#pragma unroll
    for (int i = 0; i < 8; ++i) {
        if (mask) acc[i] = 0.0f;
        outp[d0 + i] = acc[i];
        ss += acc[i] * acc[i];
    }
    // wave32 reduction
#pragma unroll
    for (int off = 16; off > 0; off >>= 1) ss += __shfl_xor(ss, off, 32);
    if (lane == 0) {
        const float r = 1.0f / (sqrtf(ss) + 1e-8f);
        if (row < NS_) rnx[row] = r; else rny[row - NS_] = r;
    }
}

// ---------------------------------------------------------------------------
// Kernel 2: WMMA GEMM  wdot = word_repr (512x256) * unit_repr^T (256x64),
// fused with multiply-only cosine normalization:
//   cosw[row,col] = 0.5 * (1 - dot * rnx[row] * rny[col])
// Tiling: block = 8 waves; wave (mt,nt) owns one 16x16 output tile.
// K loop: 64 x V_WMMA_F32_16X16X4_F32.
// A layout (16x4 f32): lane m=lane&15; VGPR0/1 = K=(lane<16?0:2)+{0,1}
// B layout (4x16 f32): lane n=lane&15; VGPR0/1 = K=(lane<16?0:2)+{0,1},
//   with B[k][n] = unit_repr[n][k].
// C/D layout: VGPR j -> rows m0+j (lanes 0-15) and m0+8+j (lanes 16-31).
// ---------------------------------------------------------------------------
__global__ void cos_kernel(const float* __restrict__ word_repr,
                           const float* __restrict__ unit_repr,
                           const float* __restrict__ rnx,
                           const float* __restrict__ rny,
                           float* __restrict__ cosw)            // NS*U
{
    const int lane = threadIdx.x & 31;
    const int wave = threadIdx.x >> 5;
    const int m0   = blockIdx.x * 32 + (wave >> 2) * 16;
    const int n0   = (wave & 3) * 16;
    const int half = lane >> 4;       // 0: K lo pair, 1: K hi pair
    const int l15  = lane & 15;
    const int kk   = half * 2;

    const float* ap = word_repr + (m0 + l15) * D_ + kk;
    const float* bp = unit_repr + (n0 + l15) * D_ + kk;

    v8f c = {};
#pragma unroll 8
    for (int k0 = 0; k0 < D_; k0 += 4) {
        v2f a, b;
        a[0] = ap[k0];     a[1] = ap[k0 + 1];
        b[0] = bp[k0];     b[1] = bp[k0 + 1];
        c = __builtin_amdgcn_wmma_f32_16x16x4_f32(
                /*neg_a=*/false, a, /*neg_b=*/false, b,
                /*c_mod=*/(short)0, c, /*reuse_a=*/false, /*reuse_b=*/false);
    }

    const int   col  = n0 + l15;
    const float rnyc = rny[col];
#pragma unroll
    for (int j = 0; j < 8; ++j) {
        const int   row = m0 + j + half * 8;
        const float dot = c[j];
        cosw[row * U_ + col] = 0.5f * (1.0f - dot * rnx[row] * rnyc);
    }
}

// ---------------------------------------------------------------------------
// Kernel 3: banded edit-distance DP. One block per sequence ns; the 10x64 cos
// slice lives in LDS; each thread strides over the 8000 tokens with the full
// DP state in registers (ls/lt loops fully unrolled -> static indices).
// Produces best_value[ns][e] = min over tokens of DP cost at ls = 4+e.
// ---------------------------------------------------------------------------
__global__ void dp_kernel(const float* __restrict__ cosw,   // NS*U
                          const int* __restrict__ segs,     // NT*MTL
                          const int* __restrict__ vlen,     // NT
                          float* __restrict__ best_value)   // NS*LEN_E
{
    __shared__ float cs[MSL_ * U_];
    __shared__ float wred[8 * LEN_E_];

    const int ns  = blockIdx.x;
    const int b   = ns >> 6, l = ns & 63;
    const int tid = threadIdx.x;

    for (int i = tid; i < MSL_ * U_; i += 256) {
        const int s = i >> 6, u = i & 63;
        int wp = l + s; if (wp > L_ - 1) wp = L_ - 1;
        cs[i] = cosw[(b * L_ + wp) * U_ + u];
    }
    __syncthreads();

    float minv[LEN_E_];
#pragma unroll
    for (int e = 0; e < LEN_E_; ++e) minv[e] = FLT_MAX;

    for (int t = tid; t < NT_; t += 256) {
        __builtin_prefetch(segs + (t + 256) * MTL_, 0, 1);  // global_prefetch_b8

        int seg[MTL_];
#pragma unroll
        for (int j = 0; j < MTL_; ++j) seg[j] = segs[t * MTL_ + j];
        const int vl = vlen[t];

        float prev[MTL_ + 1], cur[MTL_ + 1];
#pragma unroll
        for (int j = 0; j <= MTL_; ++j) prev[j] = (float)j;

#pragma unroll
        for (int ls = 1; ls <= MSL_; ++ls) {
            cur[0] = (float)ls;
#pragma unroll
            for (int j = 1; j <= MTL_; ++j) cur[j] = BIG_;
            const int lo = (ls - 2 > 1) ? (ls - 2) : 1;
            const int hi = (ls + 1 < MTL_) ? (ls + 1) : MTL_;
#pragma unroll
            for (int lt = lo; lt <= hi; ++lt) {
                const float diff = cs[(ls - 1) * U_ + seg[lt - 1]];
                const float v1 = prev[lt] + 1.0f;
                const float v2 = cur[lt - 1] + 1.0f;
                const float v3 = prev[lt - 1] + diff;
                float m = v1 < v2 ? v1 : v2;
                cur[lt] = m < v3 ? m : v3;
            }
#pragma unroll
            for (int j = 0; j <= MTL_; ++j) prev[j] = cur[j];

            if (ls >= MIN_WL_) {
                float v = BIG_;
#pragma unroll
                for (int j = MIN_WL_; j <= MTL_; ++j)
                    if (vl == j) v = cur[j];
                const int e = ls - MIN_WL_;
                minv[e] = minv[e] < v ? minv[e] : v;
            }
        }
    }

    // wave32 min-reduce, then cross-wave via LDS
    const int lane = tid & 31, wave = tid >> 5;
#pragma unroll
    for (int e = 0; e < LEN_E_; ++e) {
        float v = minv[e];
#pragma unroll
        for (int off = 16; off > 0; off >>= 1) {
            const float o = __shfl_xor(v, off, 32);
            v = v < o ? v : o;
        }
        if (lane == 0) wred[wave * LEN_E_ + e] = v;
    }
    __syncthreads();
    if (tid < LEN_E_) {
        float v = wred[tid];
#pragma unroll
        for (int w = 1; w < 8; ++w) {
            const float o = wred[w * LEN_E_ + tid];
            v = v < o ? v : o;
        }
        best_value[ns * LEN_E_ + tid] = v;
    }
}

// ---------------------------------------------------------------------------
// Kernel 4: per-batch scoring + first-occurrence argmax + matched_any.
// One block per batch; 448 live entries; lexicographic (score, -idx) reduce
// matches jnp.argmax's first-max semantics.
// ---------------------------------------------------------------------------
__global__ void final_kernel(const float* __restrict__ best_value, // NS*LEN_E
                             const int* __restrict__ lengths,      // B
                             float* __restrict__ out)              // 4*B floats
{
    __shared__ float sscore[512];
    __shared__ int   sidx[512];
    __shared__ int   smat[512];

    const int b   = blockIdx.x;
    const int tid = threadIdx.x;

    float score = -FLT_MAX;
    int   idx   = 0x7FFFFFFF;
    int   mat   = 0;
    if (tid < L_ * LEN_E_) {
        const int l = tid / LEN_E_, e = tid % LEN_E_;
        const float bv  = best_value[(b * L_ + l) * LEN_E_ + e];
        const int   len = MIN_WL_ + e;
        const int   end = l + len - 1;
        const bool  viable = end < lengths[b];
        score = viable ? (float)len * (1.0f - bv) : 0.0f;
        mat   = (viable && bv < THRESH_) ? 1 : 0;
        idx   = tid;
    }
    sscore[tid] = score; sidx[tid] = idx; smat[tid] = mat;
    __syncthreads();

    for (int off = 256; off > 0; off >>= 1) {
        if (tid < off) {
            const float s1 = sscore[tid],       s2 = sscore[tid + off];
            const int   i1 = sidx[tid],         i2 = sidx[tid + off];
            if (s2 > s1 || (s2 == s1 && i2 < i1)) { sscore[tid] = s2; sidx[tid] = i2; }
            smat[tid] |= smat[tid + off];
        }
        __syncthreads();
    }

    if (tid == 0) {
        const int bi     = sidx[0];
        const int bstart = bi / LEN_E_;
        const int bend   = bi % LEN_E_ + bstart + MIN_WL_ - 1;
        out[b]           = sscore[0];
        out[B_ + b]      = (float)bstart;
        out[2 * B_ + b]  = (float)bend;
        out[3 * B_ + b]  = smat[0] ? 1.0f : 0.0f;
    }
}

// ---------------------------------------------------------------------------
extern "C" void kernel_launch(void* const* d_in, const int* in_sizes, int n_in,
                              void* d_out, int out_size, void* d_ws, size_t ws_size,
                              hipStream_t stream) {
    const float* emb     = (const float*)d_in[0];  // NF*D
    const int*   feat    = (const int*)  d_in[1];  // B*L*G
    const int*   lengths = (const int*)  d_in[2];  // B
    const int*   ufeat   = (const int*)  d_in[3];  // U*G
    const int*   segs    = (const int*)  d_in[4];  // NT*MTL
    const int*   vlen    = (const int*)  d_in[5];  // NT
    float*       out     = (float*)d_out;

    float* ws        = (float*)d_ws;
    float* cosw      = ws;                              // NS*U      = 32768
    float* word_repr = cosw + (size_t)NS_ * U_;         // NS*D      = 131072
    float* unit_repr = word_repr + (size_t)NS_ * D_;    // U*D       = 16384
    float* rnx       = unit_repr + (size_t)U_ * D_;     // NS        = 512
    float* rny       = rnx + NS_;                       // U         = 64
    float* bval      = rny + U_;                        // NS*LEN_E  = 3584

    // 1) representations + reciprocal norms: 576 rows, 8 waves/block
    repr_kernel<<<(NS_ + U_) / 8, 256, 0, stream>>>(
        emb, feat, lengths, ufeat, word_repr, unit_repr, rnx, rny);

    // 2) WMMA GEMM + cosine: 512 rows / (32 rows per block) = 16 blocks
    cos_kernel<<<NS_ / 32, 256, 0, stream>>>(word_repr, unit_repr, rnx, rny, cosw);

    // 3) DP: one block per sequence
    dp_kernel<<<NS_, 256, 0, stream>>>(cosw, segs, vlen, bval);

    // 4) per-batch argmax / matched_any
    final_kernel<<<B_, 512, 0, stream>>>(bval, lengths, out);
}